// Kernel_13691307393279587467_48507360641595
// MI455X (gfx1250) — compile-verified
//
#include <hip/hip_runtime.h>
#include <math.h>

typedef __attribute__((ext_vector_type(2))) float v2f;
typedef __attribute__((ext_vector_type(4))) float v4f;
typedef __attribute__((ext_vector_type(8))) float v8f;
typedef __attribute__((ext_vector_type(4))) unsigned int v4u;
typedef __attribute__((ext_vector_type(4))) int v4i;
typedef __attribute__((ext_vector_type(8))) int v8i;

#define CC   384
#define HH   56
#define HW   3136      // 56*56
#define NB   32

// ---------------------------------------------------------------------------
// Fused 1x1-conv GEMM: OUT[n,m,p] = epilogue( sum_k W[m,k] * X[n,k,p] )
// MODE 0: exact-erf GELU        (t1 -> t2)
// MODE 1: multiply by AUX[idx]  (t7 * t2 -> t8)
// MODE 2: add AUX[idx]          (x + t9 -> out)
// Block: 256 threads = 8 waves; block tile 64(M) x 64(P).
// Each wave owns TWO 16x16 accumulators stacked in M (32M x 16P) so every
// global B fragment feeds two v_wmma_f32_16x16x4_f32.
// A (weights, 64 rows x 384 K = 96KB) staged in LDS via the Tensor Data
// Mover (tensor_load_to_lds + s_wait_tensorcnt), cooperative copy fallback.
// ---------------------------------------------------------------------------
template <int MODE>
__global__ __launch_bounds__(256)
void gemm1x1_wmma(const float* __restrict__ X, const float* __restrict__ W,
                  const float* AUX, float* OUT)
{
    __shared__ float As[64 * CC];   // 96 KB

    const int tid = threadIdx.x;
    const int bx = blockIdx.x;   // pixel tile (64 pixels)
    const int by = blockIdx.y;   // M tile (64 rows)
    const int bz = blockIdx.z;   // image
    const int m0 = by * 64;

    const float* Wt = W + (size_t)m0 * CC;

#if __has_builtin(__builtin_amdgcn_tensor_load_to_lds) && defined(__AMDGCN__)
    // --- TDM stage of A tile: contiguous 64x384 f32 = 24576 elements -------
    if (tid < 32) {
        unsigned lds_off = (unsigned)(size_t)(&As[0]);
        unsigned long long ga = (unsigned long long)(size_t)Wt;
        v4u g0;
        g0.x = 1u;                                   // count=1, user desc
        g0.y = lds_off;                              // lds_addr [63:32]
        g0.z = (unsigned)(ga & 0xFFFFFFFFu);         // global_addr lo
        g0.w = (unsigned)((ga >> 32) & 0x01FFFFFFu)  // global_addr hi (57b)
               | (2u << 30);                         // type=2 ("image")
        v8i g1;
        g1[0] = (int)(2u << 16);                     // data_size=2 -> 4B
        g1[1] = (int)(24576u << 16);                 // tensor_dim0 lo16 [63:48]
        g1[2] = (int)(1u << 16);                     // dim0 hi16=0, tensor_dim1=1
        g1[3] = (int)(24576u << 16);                 // tile_dim0 [127:112]
        g1[4] = 0;                                   // tile_dim1/2 unused
        g1[5] = 24576;                               // tensor_dim0_stride lo32
        g1[6] = 0;
        g1[7] = 0;
        v4i gz4 = {0, 0, 0, 0};
        v8i gz8 = {0, 0, 0, 0, 0, 0, 0, 0};
        __builtin_amdgcn_tensor_load_to_lds(g0, g1, gz4, gz4, gz8, 0);
        __builtin_amdgcn_s_wait_tensorcnt(0);
    }
#else
    // --- fallback: cooperative contiguous copy -----------------------------
#pragma unroll
    for (int it = 0; it < 24; ++it) {
        int idx = it * 1024 + tid * 4;
        *(v4f*)(&As[idx]) = *(const v4f*)(&Wt[idx]);
    }
#endif
    __syncthreads();

    const int lane = tid & 31;
    const int wave = tid >> 5;
    const int wm   = wave >> 2;           // 0..1 : 32-row M sub-tile
    const int wp   = wave & 3;            // 0..3 : P sub-tile
    const int half = lane >> 4;           // 0: K=k,k+1   1: K=k+2,k+3
    const int l16  = lane & 15;

    const int p = bx * 64 + wp * 16 + l16;            // global pixel (< 3136)
    const float* Xn = X + (size_t)bz * CC * HW;
    const float* arow0 = &As[(wm * 32 + l16) * CC];
    const float* arow1 = &As[(wm * 32 + 16 + l16) * CC];

    __builtin_prefetch(Xn + p, 0, 0);                 // global_prefetch_b8

    v8f acc0 = {};
    v8f acc1 = {};
#pragma unroll 8
    for (int k = 0; k < CC; k += 4) {
        v2f a0 = *(const v2f*)(&arow0[k + half * 2]);
        v2f a1 = *(const v2f*)(&arow1[k + half * 2]);
        const float* bp = Xn + (size_t)(k + half * 2) * HW + p;
        v2f b = { bp[0], bp[HW] };
        acc0 = __builtin_amdgcn_wmma_f32_16x16x4_f32(
            false, a0, false, b, (short)0, acc0, false, false);
        acc1 = __builtin_amdgcn_wmma_f32_16x16x4_f32(
            false, a1, false, b, (short)0, acc1, false, false);
    }

    // Epilogue + store: half 0 -> M rows v, half 1 -> M rows 8+v
    const size_t base0 =
        ((size_t)bz * CC + m0 + wm * 32 + half * 8) * HW + p;
    const size_t base1 = base0 + (size_t)16 * HW;
#pragma unroll
    for (int v = 0; v < 8; ++v) {
        size_t i0 = base0 + (size_t)v * HW;
        size_t i1 = base1 + (size_t)v * HW;
        float r0 = acc0[v];
        float r1 = acc1[v];
        if constexpr (MODE == 0) {
            r0 = 0.5f * r0 * (1.0f + erff(r0 * 0.7071067811865476f));
            r1 = 0.5f * r1 * (1.0f + erff(r1 * 0.7071067811865476f));
        } else if constexpr (MODE == 1) {
            r0 *= AUX[i0];
            r1 *= AUX[i1];
        } else {
            r0 += AUX[i0];
            r1 += AUX[i1];
        }
        OUT[i0] = r0;
        OUT[i1] = r1;
    }
}

// ---------------------------------------------------------------------------
// Depthwise 5x5, pad 2. One 16x16 output tile per block, 20x20 LDS halo tile.
// ---------------------------------------------------------------------------
__global__ __launch_bounds__(256)
void dw5_kernel(const float* __restrict__ in, const float* __restrict__ wt,
                float* __restrict__ out)
{
    __shared__ float tile[20 * 20];
    const int t   = blockIdx.x;           // 0..15 (4x4 tiles over 56x56)
    const int c   = blockIdx.y;
    const int n   = blockIdx.z;
    const int ty0 = (t >> 2) * 16, tx0 = (t & 3) * 16;
    const int tid = threadIdx.x;

    const float* src = in + ((size_t)n * CC + c) * HW;
    __builtin_prefetch(src + ty0 * HH + tx0, 0, 0);

    for (int i = tid; i < 400; i += 256) {
        int ly = i / 20, lx = i % 20;
        int gy = ty0 + ly - 2, gx = tx0 + lx - 2;
        float v = 0.0f;
        if (gy >= 0 && gy < HH && gx >= 0 && gx < HH) v = src[gy * HH + gx];
        tile[i] = v;
    }
    __syncthreads();

    const int ox = tid & 15, oy = tid >> 4;
    const int gy = ty0 + oy, gx = tx0 + ox;
    const float* wc = wt + c * 25;   // uniform -> scalar loads
    float sum = 0.0f;
#pragma unroll
    for (int i = 0; i < 5; ++i)
#pragma unroll
        for (int j = 0; j < 5; ++j)
            sum = fmaf(tile[(oy + i) * 20 + ox + j], wc[i * 5 + j], sum);

    if (gy < HH && gx < HH)
        out[((size_t)n * CC + c) * HW + gy * HH + gx] = sum;
}

// ---------------------------------------------------------------------------
// Depthwise 7x7, dilation 3, pad 9. 16x16 tile, 34x34 LDS halo tile.
// ---------------------------------------------------------------------------
__global__ __launch_bounds__(256)
void dw7d3_kernel(const float* __restrict__ in, const float* __restrict__ wt,
                  float* __restrict__ out)
{
    __shared__ float tile[34 * 34];
    const int t   = blockIdx.x;
    const int c   = blockIdx.y;
    const int n   = blockIdx.z;
    const int ty0 = (t >> 2) * 16, tx0 = (t & 3) * 16;
    const int tid = threadIdx.x;

    const float* src = in + ((size_t)n * CC + c) * HW;
    __builtin_prefetch(src + ty0 * HH + tx0, 0, 0);

    for (int i = tid; i < 34 * 34; i += 256) {
        int ly = i / 34, lx = i % 34;
        int gy = ty0 + ly - 9, gx = tx0 + lx - 9;
        float v = 0.0f;
        if (gy >= 0 && gy < HH && gx >= 0 && gx < HH) v = src[gy * HH + gx];
        tile[i] = v;
    }
    __syncthreads();

    const int ox = tid & 15, oy = tid >> 4;
    const int gy = ty0 + oy, gx = tx0 + ox;
    const float* wc = wt + c * 49;
    float sum = 0.0f;
#pragma unroll
    for (int i = 0; i < 7; ++i)
#pragma unroll
        for (int j = 0; j < 7; ++j)
            sum = fmaf(tile[(oy + i * 3) * 34 + ox + j * 3], wc[i * 7 + j], sum);

    if (gy < HH && gx < HH)
        out[((size_t)n * CC + c) * HW + gy * HH + gx] = sum;
}

// ---------------------------------------------------------------------------
extern "C" void kernel_launch(void* const* d_in, const int* in_sizes, int n_in,
                              void* d_out, int out_size, void* d_ws, size_t ws_size,
                              hipStream_t stream)
{
    const float* x  = (const float*)d_in[0];
    const float* w1 = (const float*)d_in[1];
    const float* w5 = (const float*)d_in[2];
    const float* w6 = (const float*)d_in[3];
    const float* w7 = (const float*)d_in[4];
    const float* w9 = (const float*)d_in[5];
    float* out = (float*)d_out;

    const size_t S = (size_t)NB * CC * HW;   // elements per tensor
    float* ws0 = (float*)d_ws;               // t2, later t8 (in-place epilogue ok)
    float* ws1 = ws0 + S;                    // t6
    float* t2 = ws0;
    float* t5 = out;                         // d_out doubles as t5 scratch
    float* t6 = ws1;
    float* t8 = ws0;                         // overwrite t2 element-wise (same idx)

    dim3 gblk(256);
    dim3 ggrid(HW / 64, CC / 64, NB);        // 49 x 6 x 32
    dim3 dgrid(16, CC, NB);                  // 4x4 tiles x channel x image

    // t2 = gelu(W1 @ x)
    gemm1x1_wmma<0><<<ggrid, gblk, 0, stream>>>(x, w1, nullptr, t2);
    // t5 = dw5(t2)
    dw5_kernel<<<dgrid, gblk, 0, stream>>>(t2, w5, t5);
    // t6 = dw7_dil3(t5)
    dw7d3_kernel<<<dgrid, gblk, 0, stream>>>(t5, w6, t6);
    // t8 = (W7 @ t6) * t2
    gemm1x1_wmma<1><<<ggrid, gblk, 0, stream>>>(t6, w7, t2, t8);
    // out = x + (W9 @ t8)
    gemm1x1_wmma<2><<<ggrid, gblk, 0, stream>>>(t8, w9, x, out);
}